// TaskLikelihoodLoss_71957882077474
// MI455X (gfx1250) — compile-verified
//
#include <hip/hip_runtime.h>

// Problem constants (from reference setup_inputs)
#define B_   64
#define C_   48
#define T_   4096
#define CHK  16          // chunk length for the WMMA-based blocked prefix scan
#define NCH  (T_ / CHK)  // 256 chunks per batch
#define EPSV 1e-8f

typedef float v2f __attribute__((ext_vector_type(2)));
typedef float v8f __attribute__((ext_vector_type(8)));

// ---------------------------------------------------------------------------
// Kernel 1: seq[b,t] = argmax_c actions_label[b,c,t]
// actions_label is (B, C, T): at fixed c, consecutive t are consecutive in
// memory, so consecutive threads (consecutive t) make fully coalesced loads.
// This kernel reads the 48 MB input once -> the bandwidth floor of the job.
// ---------------------------------------------------------------------------
__global__ void k_argmax(const float* __restrict__ act, int* __restrict__ seq) {
    int id = blockIdx.x * blockDim.x + threadIdx.x;
    if (id >= B_ * T_) return;
    int b = id / T_;
    int t = id - b * T_;
    const float* p = act + (size_t)b * C_ * T_ + t;
    float best = p[0];
    int arg = 0;
    #pragma unroll 4
    for (int c = 1; c < C_; ++c) {
        float v = p[(size_t)c * T_];
        if (v > best) { best = v; arg = c; }   // strict > keeps first index (jnp.argmax)
    }
    seq[id] = arg;
}

// ---------------------------------------------------------------------------
// Kernel 2: first_occ[b,r] = min{ t : seq[b,t]==r }  (T_ if never seen)
// "r unseen before position i"  <=>  first_occ[r] >= i
// ---------------------------------------------------------------------------
__global__ void k_firstocc(const int* __restrict__ seq, int* __restrict__ focc) {
    __shared__ int f[C_];
    int b = blockIdx.x;
    int tid = threadIdx.x;                 // 64 threads
    if (tid < C_) f[tid] = T_;
    __syncthreads();
    for (int t = tid; t < T_; t += 64) atomicMin(&f[seq[b * T_ + t]], t);
    __syncthreads();
    if (tid < C_) focc[b * C_ + tid] = f[tid];
}

// ---------------------------------------------------------------------------
// Kernel 3: per-chunk column sums  colsum[b,ch,r] = sum_{t in chunk} A[b,r,seq[t]]
// A (9 KB per batch, 0.6 MB total) is fully resident in the 192 MB L2.
// ---------------------------------------------------------------------------
__global__ void k_colsum(const float* __restrict__ A, const int* __restrict__ seq,
                         float* __restrict__ colsum) {
    int bc = blockIdx.x;                   // b*NCH + ch
    int b  = bc / NCH;
    int ch = bc - b * NCH;
    __shared__ int s[CHK];
    if (threadIdx.x < CHK) s[threadIdx.x] = seq[b * T_ + ch * CHK + threadIdx.x];
    __syncthreads();
    int r = threadIdx.x;                   // 64 threads, 48 active
    if (r < C_) {
        const float* Ar = A + (size_t)b * C_ * C_ + (size_t)r * C_;
        float sum = 0.f;
        #pragma unroll
        for (int t = 0; t < CHK; ++t) sum += Ar[s[t]];
        colsum[(size_t)bc * C_ + r] = sum;
    }
}

// ---------------------------------------------------------------------------
// Kernel 4: in-place exclusive scan over chunks for each (b,r):
// colsum[b,ch,r] -> carry[b,ch,r] = sum over chunks < ch. 3072 independent
// threads, 256 steps each.
// ---------------------------------------------------------------------------
__global__ void k_scan(float* __restrict__ colsum) {
    int id = blockIdx.x * blockDim.x + threadIdx.x;
    if (id >= B_ * C_) return;
    int b = id / C_;
    int r = id - b * C_;
    float run = 0.f;
    for (int ch = 0; ch < NCH; ++ch) {
        size_t idx = ((size_t)b * NCH + ch) * C_ + r;
        float v = colsum[idx];
        colsum[idx] = run;
        run += v;
    }
}

// ---------------------------------------------------------------------------
// Kernel 5 (main, WMMA): one wave32 block per (b, chunk) -> 16384 blocks.
//   Col(16x48)[t,r]  = A[b, r, seq[t]]           (gather from LDS-resident A)
//   S_local(16x48)   = L_strict(16x16) @ Col     (V_WMMA_F32_16X16X4_F32 x12)
//   S[i,r]           = carry[b,ch,r] + S_local[i,r]
//   num = S[i,seq[i]];  den = sum_{first_occ[r]>=gi} S[i,r]
//   term = log(max+eps) - log(min+eps), masked by (1<=gi<=T-2, seq[i]!=seq[i-1])
// ---------------------------------------------------------------------------
#define ASTR 49   // padded LDS row strides (odd -> conflict-free strided reads)
#define CSTR 49
#define SSTR 49

__global__ void __launch_bounds__(32)
k_main(const float* __restrict__ A, const int* __restrict__ seq,
       const int* __restrict__ focc, const float* __restrict__ carry,
       float* __restrict__ partial) {
    __shared__ float A_lds[C_ * ASTR];
    __shared__ float Col_lds[CHK * CSTR];
    __shared__ float S_lds[CHK * SSTR];
    __shared__ float carry_lds[C_];
    __shared__ int   focc_lds[C_];
    __shared__ int   seq_ch[CHK + 1];      // seq[gi0-1 .. gi0+15]

    int bc = blockIdx.x;                   // b*NCH + ch
    int b  = bc / NCH;
    int ch = bc - b * NCH;
    int lane = threadIdx.x;                // wave32, all 32 lanes active (EXEC all-1s for WMMA)
    int gi0 = ch * CHK;

    // ---- stage LDS: A[b] (coalesced), carry, first_occ, seq window ----
    const float* Ab = A + (size_t)b * C_ * C_;
    for (int idx = lane; idx < C_ * C_; idx += 32)
        A_lds[(idx / C_) * ASTR + (idx % C_)] = Ab[idx];
    carry_lds[lane] = carry[(size_t)bc * C_ + lane];
    focc_lds[lane]  = focc[b * C_ + lane];
    if (lane < 16) {
        carry_lds[lane + 32] = carry[(size_t)bc * C_ + lane + 32];
        focc_lds[lane + 32]  = focc[b * C_ + lane + 32];
    }
    if (lane < CHK + 1) {
        int g = gi0 - 1 + lane;
        seq_ch[lane] = (g >= 0) ? seq[b * T_ + g] : 0;
    }
    __syncthreads();

    // ---- build Col tile: Col[t,r] = A[b, r, seq[gi0+t]] ----
    for (int idx = lane; idx < CHK * C_; idx += 32) {
        int t = idx / C_;
        int r = idx - t * C_;
        Col_lds[t * CSTR + r] = A_lds[r * ASTR + seq_ch[t + 1]];
    }
    __syncthreads();

    // ---- S_local = L_strict @ Col via chained f32 WMMA (K=16 -> 4 steps) ----
    // A frag (16x4 f32): lane -> M = lane&15, VGPR v -> K = 2*(lane>=16)+v
    // B frag (4x16 f32): lane -> N = lane&15, VGPR v -> K = 2*(lane>=16)+v
    // C/D   (16x16 f32): lane -> N = lane&15, VGPR v -> M = 8*(lane>=16)+v
    int n  = lane & 15;
    int hi = lane >> 4;
    int kb = hi * 2;
    #pragma unroll
    for (int nt = 0; nt < 3; ++nt) {
        int rb = nt * 16 + n;
        v8f acc = {};
        #pragma unroll
        for (int k = 0; k < 4; ++k) {
            int k0 = 4 * k + kb;
            v2f af, bf;
            af.x = (k0     < n) ? 1.0f : 0.0f;   // L_strict[m, k0]   (m = lane&15 = n)
            af.y = (k0 + 1 < n) ? 1.0f : 0.0f;   // L_strict[m, k0+1]
            bf.x = Col_lds[ k0      * CSTR + rb];
            bf.y = Col_lds[(k0 + 1) * CSTR + rb];
            acc = __builtin_amdgcn_wmma_f32_16x16x4_f32(
                      false, af, false, bf, (short)0, acc, false, false);
        }
        #pragma unroll
        for (int v = 0; v < 8; ++v)
            S_lds[(v + 8 * hi) * SSTR + rb] = acc[v];
    }
    __syncthreads();

    // ---- per-position term: num/den from S = carry + S_local ----
    float acc_term = 0.f;                  // meaningful on lane 0
    for (int i = 0; i < CHK; ++i) {
        int gi   = gi0 + i;
        int s_i  = seq_ch[i + 1];
        int prev = seq_ch[i];
        // wave-uniform predicate (depends only on scalars)
        if (gi >= 1 && gi <= T_ - 2 && s_i != prev) {
            float p = 0.f;
            {
                float sv = carry_lds[lane] + S_lds[i * SSTR + lane];
                if (focc_lds[lane] >= gi) p += sv;
            }
            if (lane < 16) {
                int r2 = lane + 32;
                float sv = carry_lds[r2] + S_lds[i * SSTR + r2];
                if (focc_lds[r2] >= gi) p += sv;
            }
            #pragma unroll
            for (int off = 16; off >= 1; off >>= 1)
                p += __shfl_xor(p, off, 32);
            if (lane == 0) {
                float num = carry_lds[s_i] + S_lds[i * SSTR + s_i];
                float den = p;
                float d_ = fmaxf(den, num);
                float n_ = fminf(num, den);
                acc_term += logf(d_ + EPSV) - logf(n_ + EPSV);
            }
        }
    }
    if (lane == 0) partial[bc] = acc_term;
}

// ---------------------------------------------------------------------------
// Kernel 6: deterministic reduction of the 16384 per-chunk partials.
// Single block, fixed summation order -> bit-identical across graph replays
// (no float atomics anywhere in the pipeline).
// ---------------------------------------------------------------------------
__global__ void k_reduce(const float* __restrict__ partial, float* __restrict__ out) {
    __shared__ float sm[256];
    float s = 0.f;
    for (int i = threadIdx.x; i < B_ * NCH; i += 256) s += partial[i];
    sm[threadIdx.x] = s;
    __syncthreads();
    for (int off = 128; off >= 1; off >>= 1) {
        if (threadIdx.x < off) sm[threadIdx.x] += sm[threadIdx.x + off];
        __syncthreads();
    }
    if (threadIdx.x == 0) out[0] = sm[0];
}

// ---------------------------------------------------------------------------
// Launch. Workspace layout (≈4.1 MB total):
//   [0)            seq     B*T int
//   [+1 MB)        focc    B*48 int
//   [+12 KB)       carry   B*NCH*48 float  (colsum, scanned in place)
//   [+3 MB)        partial B*NCH float
// ---------------------------------------------------------------------------
extern "C" void kernel_launch(void* const* d_in, const int* in_sizes, int n_in,
                              void* d_out, int out_size, void* d_ws, size_t ws_size,
                              hipStream_t stream) {
    (void)in_sizes; (void)n_in; (void)out_size; (void)ws_size;
    const float* A   = (const float*)d_in[0];   // (B, C, C) f32
    const float* act = (const float*)d_in[1];   // (B, C, T) f32
    float* out = (float*)d_out;                 // scalar f32

    char* w = (char*)d_ws;
    size_t off = 0;
    int*   seq     = (int*)(w + off);  off += (size_t)B_ * T_ * sizeof(int);
    int*   focc    = (int*)(w + off);  off += (size_t)B_ * C_ * sizeof(int);
    float* carry   = (float*)(w + off); off += (size_t)B_ * NCH * C_ * sizeof(float);
    float* partial = (float*)(w + off); off += (size_t)B_ * NCH * sizeof(float);

    k_argmax  <<<(B_ * T_ + 255) / 256, 256, 0, stream>>>(act, seq);
    k_firstocc<<<B_, 64, 0, stream>>>(seq, focc);
    k_colsum  <<<B_ * NCH, 64, 0, stream>>>(A, seq, carry);
    k_scan    <<<(B_ * C_ + 255) / 256, 256, 0, stream>>>(carry);
    k_main    <<<B_ * NCH, 32, 0, stream>>>(A, seq, focc, carry, partial);
    k_reduce  <<<1, 256, 0, stream>>>(partial, out);
}